// GAT_60902636257639
// MI455X (gfx1250) — compile-verified
//
#include <hip/hip_runtime.h>
#include <math.h>

#define HEADS 4
#define FEAT  32
#define HF    128   // HEADS*FEAT

typedef __attribute__((ext_vector_type(2))) float v2f;
typedef __attribute__((ext_vector_type(8))) float v8f;

// ------------------------------------------------------------------
// WMMA f32 GEMM: Hout[M,128] = X[M,K] @ W[K,128]
// block = 256 threads = 8 waves; wave w computes 16x16 tile at col w*16.
// Uses V_WMMA_F32_16X16X4_F32 (full fp32 precision, CDNA5 matrix pipe).
// ------------------------------------------------------------------
__global__ void __launch_bounds__(256)
gat_gemm_wmma_f32(const float* __restrict__ X, const float* __restrict__ W,
                  float* __restrict__ Hout, int K) {
  const int wave = threadIdx.x >> 5;
  const int lane = threadIdx.x & 31;
  const int half = lane >> 4;          // 0: K={0,1}, 1: K={2,3}
  const int l15  = lane & 15;
  const int row  = blockIdx.x * 16 + l15;   // A row for this lane
  const int col  = wave * 16 + l15;         // B/C col for this lane
  const int kh   = half * 2;
  const float* __restrict__ xrow = X + (size_t)row * K;

  v8f c = {};
  for (int k0 = 0; k0 < K; k0 += 4) {
    v2f a, b;
    a.x = xrow[k0 + kh];
    a.y = xrow[k0 + kh + 1];
    b.x = W[(size_t)(k0 + kh)     * HF + col];
    b.y = W[(size_t)(k0 + kh + 1) * HF + col];
    // 8 args: (neg_a, A, neg_b, B, c_mod, C, reuse_a, reuse_b)
    c = __builtin_amdgcn_wmma_f32_16x16x4_f32(false, a, false, b,
                                              (short)0, c, false, false);
  }
  // C/D layout: VGPR v -> row (v + 8*half), col = lane&15
  const int rbase = blockIdx.x * 16 + half * 8;
  float* orow = Hout + (size_t)rbase * HF + col;
#pragma unroll
  for (int v = 0; v < 8; ++v)
    orow[(size_t)v * HF] = c[v];
}

// Scalar tail GEMM for rows not covered by 16-row tiles (N%16 != 0 case).
__global__ void gat_gemm_tail(const float* __restrict__ X, const float* __restrict__ W,
                              float* __restrict__ Hout, int K, int row0, int Nn) {
  int i = blockIdx.x * blockDim.x + threadIdx.x;
  int rows = Nn - row0;
  if (i >= rows * HF) return;
  int r = row0 + (i >> 7), col = i & (HF - 1);
  float acc = 0.f;
  for (int k = 0; k < K; ++k) acc += X[(size_t)r * K + k] * W[(size_t)k * HF + col];
  Hout[(size_t)r * HF + col] = acc;
}

// ------------------------------------------------------------------
// Per-node attention coefficients: a_src[n,h] = <h[n,h,:], att_src[h,:]>
// ------------------------------------------------------------------
__global__ void gat_alpha(const float* __restrict__ H, const float* __restrict__ att_src,
                          const float* __restrict__ att_dst,
                          float* __restrict__ a_src, float* __restrict__ a_dst, int Nn) {
  int i = blockIdx.x * blockDim.x + threadIdx.x;   // n*HEADS + h
  if (i >= Nn * HEADS) return;
  int h = i & (HEADS - 1);
  const float4* hp = (const float4*)(H + (size_t)(i >> 2) * HF + h * FEAT);
  const float4* as = (const float4*)(att_src + h * FEAT);
  const float4* ad = (const float4*)(att_dst + h * FEAT);
  float s1 = 0.f, s2 = 0.f;
#pragma unroll
  for (int f = 0; f < FEAT / 4; ++f) {
    float4 v = hp[f], a = as[f], d = ad[f];
    s1 += v.x * a.x + v.y * a.y + v.z * a.z + v.w * a.w;
    s2 += v.x * d.x + v.y * d.y + v.z * d.z + v.w * d.w;
  }
  a_src[i] = s1;
  a_dst[i] = s2;
}

// ------------------------------------------------------------------
// Init accumulators: m = -inf, s = 0, acc = 0 (acc zeroed 4-wide, b128)
// ------------------------------------------------------------------
__global__ void gat_init(float* __restrict__ m, float* __restrict__ s,
                         float4* __restrict__ acc4, int Nn) {
  int i = blockIdx.x * blockDim.x + threadIdx.x;
  if (i < Nn * HEADS) { m[i] = -INFINITY; s[i] = 0.f; }
  if (i < Nn * (HF / 4)) acc4[i] = make_float4(0.f, 0.f, 0.f, 0.f);
}

__device__ __forceinline__ float leaky(float e) { return e > 0.f ? e : 0.2f * e; }

// float atomic-max via monotone int/uint reinterpretation
__device__ __forceinline__ void atomicMaxF(float* addr, float val) {
  if (val >= 0.f) atomicMax((int*)addr, __float_as_int(val));
  else            atomicMin((unsigned int*)addr, __float_as_uint(val));
}

__device__ __forceinline__ void edge_pair(const int* __restrict__ ei, int eid, int E,
                                          int& s, int& d) {
  if (eid < E) { s = ei[eid]; d = ei[E + eid]; }
  else         { s = d = eid - E; }          // appended self-loops
}

// ------------------------------------------------------------------
// Pass 1: scatter-max of leaky-relu logits per destination node
// ------------------------------------------------------------------
__global__ void gat_edge_max(const int* __restrict__ ei, const float* __restrict__ asrc,
                             const float* __restrict__ adst, float* __restrict__ m,
                             int E, int Nn) {
  int i = blockIdx.x * blockDim.x + threadIdx.x;
  if (i >= (E + Nn) * HEADS) return;
  int eid = i >> 2, h = i & (HEADS - 1), s, d;
  edge_pair(ei, eid, E, s, d);
  float e = leaky(asrc[s * HEADS + h] + adst[d * HEADS + h]);
  atomicMaxF(&m[d * HEADS + h], e);
}

// ------------------------------------------------------------------
// Pass 2: scatter-sum of exp(e - m[dst])
// ------------------------------------------------------------------
__global__ void gat_edge_sum(const int* __restrict__ ei, const float* __restrict__ asrc,
                             const float* __restrict__ adst, const float* __restrict__ m,
                             float* __restrict__ sums, int E, int Nn) {
  int i = blockIdx.x * blockDim.x + threadIdx.x;
  if (i >= (E + Nn) * HEADS) return;
  int eid = i >> 2, h = i & (HEADS - 1), s, d;
  edge_pair(ei, eid, E, s, d);
  float e = leaky(asrc[s * HEADS + h] + adst[d * HEADS + h]);
  atomicAdd(&sums[d * HEADS + h], expf(e - m[d * HEADS + h]));
}

// ------------------------------------------------------------------
// Pass 3: weighted aggregation acc[dst,h,f] += alpha * h[src,h,f]
// One thread per (edge, 4-channel group): b128 gather of source feats,
// one weight computation per 4 channels, 4 per-dword f32 atomics that
// land in L2 (acc is 25.6MB, whole working set < 192MB L2).
// ------------------------------------------------------------------
__global__ void gat_edge_agg(const int* __restrict__ ei, const float* __restrict__ asrc,
                             const float* __restrict__ adst, const float* __restrict__ m,
                             const float* __restrict__ sums, const float* __restrict__ hfeat,
                             float* __restrict__ acc, int E, int Nn) {
  long long i = (long long)blockIdx.x * blockDim.x + threadIdx.x;
  if (i >= (long long)(E + Nn) * (HF / 4)) return;
  int eid = (int)(i >> 5);            // HF/4 = 32 groups per edge
  int g   = (int)(i & 31);
  int c   = g * 4;                    // first channel of this group
  int h   = c >> 5;                   // head index
  int s, d;
  edge_pair(ei, eid, E, s, d);
  float e = leaky(asrc[s * HEADS + h] + adst[d * HEADS + h]);
  float w = expf(e - m[d * HEADS + h]) / (sums[d * HEADS + h] + 1e-16f);
  const float4 hv = *(const float4*)(hfeat + (size_t)s * HF + c);
  float* ap = acc + (size_t)d * HF + c;
  atomicAdd(ap + 0, w * hv.x);
  atomicAdd(ap + 1, w * hv.y);
  atomicAdd(ap + 2, w * hv.z);
  atomicAdd(ap + 3, w * hv.w);
}

// ------------------------------------------------------------------
// Head-average + bias (+ optional ELU for layer-1 output)
// ------------------------------------------------------------------
__global__ void gat_head_mean(const float* __restrict__ acc, const float* __restrict__ bias,
                              float* __restrict__ out, int Nn, int do_elu) {
  int i = blockIdx.x * blockDim.x + threadIdx.x;   // n*FEAT + f
  if (i >= Nn * FEAT) return;
  int n = i >> 5, f = i & (FEAT - 1);
  const float* a = acc + (size_t)n * HF;
  float v = 0.25f * (a[f] + a[FEAT + f] + a[2 * FEAT + f] + a[3 * FEAT + f]) + bias[f];
  if (do_elu) v = v > 0.f ? v : (expf(v) - 1.f);
  out[i] = v;
}

// ------------------------------------------------------------------
extern "C" void kernel_launch(void* const* d_in, const int* in_sizes, int n_in,
                              void* d_out, int out_size, void* d_ws, size_t ws_size,
                              hipStream_t stream) {
  const float* x   = (const float*)d_in[0];
  const int*   ei  = (const int*)d_in[1];
  const float* W1  = (const float*)d_in[2];
  const float* as1 = (const float*)d_in[3];
  const float* ad1 = (const float*)d_in[4];
  const float* b1  = (const float*)d_in[5];
  const float* W2  = (const float*)d_in[6];
  const float* as2 = (const float*)d_in[7];
  const float* ad2 = (const float*)d_in[8];
  const float* b2  = (const float*)d_in[9];

  const int Nn = in_sizes[0] / HF;   // 50000
  const int E  = in_sizes[1] / 2;    // 1600000
  const int ET = E + Nn;

  // workspace layout (floats): h | acc | a_src | a_dst | m | s | x2
  float* ws    = (float*)d_ws;
  float* hbuf  = ws;
  float* acc   = hbuf + (size_t)Nn * HF;
  float* asrc  = acc  + (size_t)Nn * HF;
  float* adst  = asrc + (size_t)Nn * HEADS;
  float* mbuf  = adst + (size_t)Nn * HEADS;
  float* sbuf  = mbuf + (size_t)Nn * HEADS;
  float* x2    = sbuf + (size_t)Nn * HEADS;

  const dim3 blk(256);
  const int row_tiles = Nn / 16;
  const int tail_row0 = row_tiles * 16;

  for (int layer = 0; layer < 2; ++layer) {
    const float* Xin = (layer == 0) ? x  : x2;
    const int    K   = (layer == 0) ? 128 : FEAT;
    const float* Wm  = (layer == 0) ? W1 : W2;
    const float* as  = (layer == 0) ? as1 : as2;
    const float* ad  = (layer == 0) ? ad1 : ad2;
    const float* bb  = (layer == 0) ? b1 : b2;
    float*       op  = (layer == 0) ? x2 : (float*)d_out;

    if (row_tiles > 0)
      gat_gemm_wmma_f32<<<row_tiles, blk, 0, stream>>>(Xin, Wm, hbuf, K);
    if (tail_row0 < Nn) {
      int tail = (Nn - tail_row0) * HF;
      gat_gemm_tail<<<(tail + 255) / 256, blk, 0, stream>>>(Xin, Wm, hbuf, K, tail_row0, Nn);
    }
    gat_alpha<<<(Nn * HEADS + 255) / 256, blk, 0, stream>>>(hbuf, as, ad, asrc, adst, Nn);
    gat_init<<<(Nn * (HF / 4) + 255) / 256, blk, 0, stream>>>(mbuf, sbuf, (float4*)acc, Nn);
    gat_edge_max<<<(ET * HEADS + 255) / 256, blk, 0, stream>>>(ei, asrc, adst, mbuf, E, Nn);
    gat_edge_sum<<<(ET * HEADS + 255) / 256, blk, 0, stream>>>(ei, asrc, adst, mbuf, sbuf, E, Nn);
    {
      long long tot = (long long)ET * (HF / 4);
      unsigned int nb = (unsigned int)((tot + 255) / 256);
      gat_edge_agg<<<nb, blk, 0, stream>>>(ei, asrc, adst, mbuf, sbuf, hbuf, acc, E, Nn);
    }
    gat_head_mean<<<(Nn * FEAT + 255) / 256, blk, 0, stream>>>(acc, bb, op, Nn, layer == 0 ? 1 : 0);
  }
}